// GaussianKDEOffsetGenerator_15556371546702
// MI455X (gfx1250) — compile-verified
//
#include <hip/hip_runtime.h>
#include <hip/hip_bf16.h>
#include <float.h>

typedef float v2f __attribute__((ext_vector_type(2)));
typedef float v8f __attribute__((ext_vector_type(8)));

#define PE_DIMC 32
#define HIDC    16
#define KPTS    32
#define WPB     8            // waves (batch-blocks) per workgroup
#define SIGMA_F 1.0f
#define EPS_F   1e-6f

// relu in a single v_med3_f32 (avoids the fmaxf sNaN-canonicalization self-max)
__device__ __forceinline__ float relu1(float x) {
    return __builtin_amdgcn_fmed3f(x, 0.0f, FLT_MAX);
}

__launch_bounds__(256)
__global__ void kde_offset_kernel(const float* __restrict__ coords,
                                  const float* __restrict__ Wpe,
                                  const float* __restrict__ bpe,
                                  const float* __restrict__ W1,
                                  const float* __restrict__ b1,
                                  const float* __restrict__ W2,
                                  const float* __restrict__ b2,
                                  float* __restrict__ out,
                                  int bn)
{
    // per-wave private LDS slabs (padded to avoid bank conflicts)
    __shared__ float sc  [WPB][KPTS * 3];
    __shared__ float peL [WPB][KPTS][PE_DIMC + 1];
    __shared__ float kdeL[WPB][KPTS];
    __shared__ float hL  [WPB][KPTS][HIDC + 1];

    const int  lane = threadIdx.x & 31;
    const int  wid  = threadIdx.x >> 5;
    const int  rawb = blockIdx.x * WPB + wid;
    const bool writer = rawb < bn;
    const int  blk  = writer ? rawb : (bn - 1);   // clamp: keep waves convergent

    const int  n  = lane & 15;
    const bool lo = lane < 16;

    // ---- coords for this block -> LDS (lane l owns point l) ----
    {
        const float* cb = coords + (size_t)blk * (KPTS * 3);
        sc[wid][lane * 3 + 0] = cb[lane * 3 + 0];
        sc[wid][lane * 3 + 1] = cb[lane * 3 + 1];
        sc[wid][lane * 3 + 2] = cb[lane * 3 + 2];
    }
    __syncthreads();

    // ---- B fragments: W_pe (3x32, K padded to 4) as two 4x16 tiles ----
    // B layout (4x16 f32): lanes 0-15 hold K=0 (x) / K=1 (y); lanes 16-31 hold K=2 (x) / K=3 (y)
    v2f bpe_n0, bpe_n1;
    bpe_n0.x = lo ? Wpe[0 * PE_DIMC + n]      : Wpe[2 * PE_DIMC + n];
    bpe_n0.y = lo ? Wpe[1 * PE_DIMC + n]      : 0.0f;
    bpe_n1.x = lo ? Wpe[0 * PE_DIMC + n + 16] : Wpe[2 * PE_DIMC + n + 16];
    bpe_n1.y = lo ? Wpe[1 * PE_DIMC + n + 16] : 0.0f;

    // bias b_pe[n] is constant across a lane's 8 C/D VGPRs (column-indexed layout):
    // feed it through the WMMA C operand instead of VALU adds after the fact.
    v8f cbias0, cbias1;
    {
        const float bias0 = bpe[n];
        const float bias1 = bpe[n + 16];
        #pragma unroll
        for (int v = 0; v < 8; ++v) { cbias0[v] = bias0; cbias1[v] = bias1; }
    }

    // ---- B fragments: W1 (32x16) as 8 K-step tiles of 4x16 ----
    v2f bw1[8];
    #pragma unroll
    for (int ks = 0; ks < 8; ++ks) {
        const int kr = ks * 4 + (lo ? 0 : 2);
        bw1[ks].x = W1[kr * HIDC + n];
        bw1[ks].y = W1[(kr + 1) * HIDC + n];
    }
    v8f cbias_h;
    {
        const float bb1 = b1[n];
        #pragma unroll
        for (int v = 0; v < 8; ++v) cbias_h[v] = bb1;
    }

    // ---- KDE: lane i sums exp(-|c_i - c_j|^2 / (2s^2+eps)) over j ----
    const float invDen = 1.0f / (2.0f * SIGMA_F * SIGMA_F + EPS_F);
    const float coef   = 0.3989422f;   // 1/sqrt(2*pi*s^2 + eps)
    {
        const float clx = sc[wid][lane * 3 + 0];
        const float cly = sc[wid][lane * 3 + 1];
        const float clz = sc[wid][lane * 3 + 2];
        float acc = 0.0f;
        #pragma unroll 8
        for (int j = 0; j < KPTS; ++j) {
            const float dx = clx - sc[wid][j * 3 + 0];
            const float dy = cly - sc[wid][j * 3 + 1];
            const float dz = clz - sc[wid][j * 3 + 2];
            const float d2 = dx * dx + dy * dy + dz * dz;
            acc += __expf(-d2 * invDen);
        }
        acc *= coef;
        float mx = acc;
        #pragma unroll
        for (int off = 16; off > 0; off >>= 1)
            mx = fmaxf(mx, __shfl_xor(mx, off, 32));
        kdeL[wid][lane] = acc / (mx + EPS_F);
    }

    // ---- PE einsum via WMMA F32 16x16x4: per i, pairs (i, j=0..31) ----
    for (int i = 0; i < KPTS; ++i) {
        const float cix = sc[wid][i * 3 + 0];
        const float ciy = sc[wid][i * 3 + 1];
        const float ciz = sc[wid][i * 3 + 2];
        float s0 = 0.0f, s1 = 0.0f;
        #pragma unroll
        for (int jb = 0; jb < KPTS; jb += 16) {
            const int j = jb + n;                 // A row m = n, same j for lanes n and n+16
            const float rx = cix - sc[wid][j * 3 + 0];
            const float ry = ciy - sc[wid][j * 3 + 1];
            const float rz = ciz - sc[wid][j * 3 + 2];
            // A layout (16x4 f32): lanes 0-15 K=0/1, lanes 16-31 K=2/3 (K=3 is zero pad)
            v2f a;
            a.x = lo ? rx : rz;
            a.y = lo ? ry : 0.0f;
            // D = A*B + bias (bias rides in the C operand)
            v8f d0 = __builtin_amdgcn_wmma_f32_16x16x4_f32(false, a, false, bpe_n0,
                                                           (short)0, cbias0, false, false);
            v8f d1 = __builtin_amdgcn_wmma_f32_16x16x4_f32(false, a, false, bpe_n1,
                                                           (short)0, cbias1, false, false);
            // relu (single v_med3) then sum over the j-rows of the D tile
            #pragma unroll
            for (int v = 0; v < 8; ++v) {
                s0 += relu1(d0[v]);
                s1 += relu1(d1[v]);
            }
        }
        // column n lives in lanes {n, n+16}: finish the j-sum across the lane halves
        s0 += __shfl_xor(s0, 16, 32);
        s1 += __shfl_xor(s1, 16, 32);
        peL[wid][i][lane] = (lo ? s0 : s1) * (1.0f / 32.0f);   // mean over j; lane d holds pe[i][d]
    }
    __syncthreads();

    // ---- h = relu(fused @ W1 + b1) via WMMA, fused = pe * kde folded on load ----
    #pragma unroll
    for (int mt = 0; mt < 2; ++mt) {
        v8f c = cbias_h;    // seed the K-accumulation with the bias
        #pragma unroll
        for (int ks = 0; ks < 8; ++ks) {
            const int   mrow = mt * 16 + n;
            const int   kk   = ks * 4 + (lo ? 0 : 2);
            const float kv   = kdeL[wid][mrow];
            v2f a;
            a.x = peL[wid][mrow][kk]     * kv;
            a.y = peL[wid][mrow][kk + 1] * kv;
            c = __builtin_amdgcn_wmma_f32_16x16x4_f32(false, a, false, bw1[ks],
                                                      (short)0, c, false, false);
        }
        #pragma unroll
        for (int v = 0; v < 8; ++v) {
            const int row = mt * 16 + v + (lo ? 0 : 8);        // D: VGPR v -> M=v (lo) / v+8 (hi)
            hL[wid][row][n] = relu1(c[v]);
        }
    }
    __syncthreads();

    // ---- delta = h @ W2 + b2 (32x16x3: scalar per lane i) ----
    {
        float o0 = b2[0], o1 = b2[1], o2 = b2[2];
        #pragma unroll
        for (int hd = 0; hd < HIDC; ++hd) {
            const float hv = hL[wid][lane][hd];
            o0 += hv * W2[hd * 3 + 0];
            o1 += hv * W2[hd * 3 + 1];
            o2 += hv * W2[hd * 3 + 2];
        }
        if (writer) {
            float* ob = out + (size_t)blk * (KPTS * 3) + lane * 3;
            ob[0] = o0;
            ob[1] = o1;
            ob[2] = o2;
        }
    }
}

extern "C" void kernel_launch(void* const* d_in, const int* in_sizes, int n_in,
                              void* d_out, int out_size, void* d_ws, size_t ws_size,
                              hipStream_t stream) {
    const float* coords = (const float*)d_in[0];
    const float* Wpe    = (const float*)d_in[1];
    const float* bpe    = (const float*)d_in[2];
    const float* W1     = (const float*)d_in[3];
    const float* b1     = (const float*)d_in[4];
    const float* W2     = (const float*)d_in[5];
    const float* b2     = (const float*)d_in[6];
    float* out = (float*)d_out;

    const int bn   = in_sizes[0] / (KPTS * 3);   // 8192
    const int grid = (bn + WPB - 1) / WPB;       // one wave per batch-block
    kde_offset_kernel<<<grid, 256, 0, stream>>>(coords, Wpe, bpe, W1, b1, W2, b2, out, bn);
}